// EdgeConditionedConv_77584289235401
// MI455X (gfx1250) — compile-verified
//
#include <hip/hip_runtime.h>

#define N_NODES 100000
#define N_EDGES 1600000
#define IN_DIM 16
#define OUT_DIM 16
#define EDGE_DIM 8

typedef __attribute__((ext_vector_type(2))) float v2f;
typedef __attribute__((ext_vector_type(8))) float v8f;

// D(16x16,f32) = A(16x4,f32) * B(4x16,f32) + C   -- V_WMMA_F32_16X16X4_F32
__device__ __forceinline__ v8f wmma4(v2f a, v2f b, v8f c) {
  return __builtin_amdgcn_wmma_f32_16x16x4_f32(
      /*neg_a=*/false, a, /*neg_b=*/false, b,
      /*c_mod=*/(short)0, c, /*reuse_a=*/false, /*reuse_b=*/false);
}

// -------- Kernel 1: out = x @ root_w + root_b  (one 16-node tile per wave iter)
__global__ void ecconv_root_kernel(const float* __restrict__ x,
                                   const float* __restrict__ root_w,
                                   const float* __restrict__ root_b,
                                   float* __restrict__ out, int numTiles) {
  const int lane = threadIdx.x & 31;
  const int col  = lane & 15;   // N index (output channel o)
  const int hi   = lane >> 4;   // upper half of wave
  const int wavesPerBlock = blockDim.x >> 5;
  const int wave   = blockIdx.x * wavesPerBlock + (threadIdx.x >> 5);
  const int stride = gridDim.x * wavesPerBlock;

  // B operand slices: value root_w[i*16 + col], i = 4s + r + 2*hi (r = vgpr 0/1)
  v2f B[4];
#pragma unroll
  for (int s = 0; s < 4; ++s) {
    int i0 = 4 * s + 2 * hi;
    B[s].x = root_w[i0 * 16 + col];
    B[s].y = root_w[(i0 + 1) * 16 + col];
  }
  const float rb = root_b[col];

  for (int tile = wave; tile < numTiles; tile += stride) {
    const float* xrow = x + (tile * 16 + col) * IN_DIM + 2 * hi;
    v8f c;
#pragma unroll
    for (int r = 0; r < 8; ++r) c[r] = rb;   // bias broadcast (same for all rows)
#pragma unroll
    for (int s = 0; s < 4; ++s) {
      v2f a = *(const v2f*)(xrow + 4 * s);    // i = 4s+2hi, 4s+2hi+1
      c = wmma4(a, B[s], c);
    }
    // C layout: row M = r + 8*hi, col = lane%16
#pragma unroll
    for (int r = 0; r < 8; ++r)
      out[(tile * 16 + r + 8 * hi) * OUT_DIM + col] = c[r];
  }
}

// -------- Kernel 2: per-edge dynamic-weight matvec + scatter-add
// Reformulated: Y[e, k*16+o] = sum_i x[src_e,i] * edge_w[k, o*16+i]   (WMMA GEMM)
//               msg[e,o] = sum_k attr[e,k]*Y[e,k,o] + sum_i x[src_e,i]*edge_b[o*16+i]
__global__ void ecconv_edge_kernel(const float* __restrict__ x,
                                   const float* __restrict__ edge_attr,
                                   const float* __restrict__ edge_w,
                                   const float* __restrict__ edge_b,
                                   const long long* __restrict__ edge_index,
                                   float* __restrict__ out, int numTiles) {
  const int lane = threadIdx.x & 31;
  const int col  = lane & 15;
  const int hi   = lane >> 4;
  const int wavesPerBlock = blockDim.x >> 5;
  const int wave   = blockIdx.x * wavesPerBlock + (threadIdx.x >> 5);
  const int stride = gridDim.x * wavesPerBlock;

  const long long* srcIdx = edge_index;             // edge_index[0,:]
  const long long* dstIdx = edge_index + N_EDGES;   // edge_index[1,:]

  // Loop-invariant B operands: tile t <-> k=t, col <-> o, i = 4s + r + 2*hi.
  // B value = edge_w[t*256 + col*16 + i]  (i-contiguous -> v2f load)
  v2f Bw[8][4];
#pragma unroll
  for (int t = 0; t < 8; ++t)
#pragma unroll
    for (int s = 0; s < 4; ++s)
      Bw[t][s] = *(const v2f*)(edge_w + t * 256 + col * 16 + 4 * s + 2 * hi);
  v2f Bb[4];
#pragma unroll
  for (int s = 0; s < 4; ++s)
    Bb[s] = *(const v2f*)(edge_b + col * 16 + 4 * s + 2 * hi);

  const int permBase = 8 * hi;  // shfl source-lane base for C-layout row r+8*hi

  for (int tile = wave; tile < numTiles; tile += stride) {
    const int e0 = tile * 16;
    const int myEdge = e0 + col;                 // lane owns edge e0 + lane%16
    const int srcn = (int)srcIdx[myEdge];
    const int dstn = (int)dstIdx[myEdge];

    // edge_attr row of "my" edge (broadcast later via ds_bpermute)
    const float4* ap = (const float4*)(edge_attr + (long)myEdge * EDGE_DIM);
    float4 a0 = ap[0], a1 = ap[1];
    float av[8] = {a0.x, a0.y, a0.z, a0.w, a1.x, a1.y, a1.z, a1.w};

    // A operand: gathered source features (L2-resident), A layout K = 4s + r + 2*hi
    const float* xrow = x + (long)srcn * IN_DIM + 2 * hi;
    v2f A[4];
#pragma unroll
    for (int s = 0; s < 4; ++s)
      A[s] = *(const v2f*)(xrow + 4 * s);

    // bias GEMM tile -> msg accumulator (C layout: e = r+8*hi, o = col)
    v8f msg = {};
#pragma unroll
    for (int s = 0; s < 4; ++s)
      msg = wmma4(A[s], Bb[s], msg);

    // 8 k-tiles: Y_t then msg += attr[e,t] * Y_t  (no cross-lane reduction)
#pragma unroll
    for (int t = 0; t < 8; ++t) {
      v8f y = {};
#pragma unroll
      for (int s = 0; s < 4; ++s)
        y = wmma4(A[s], Bw[t][s], y);
#pragma unroll
      for (int r = 0; r < 8; ++r) {
        float a = __shfl(av[t], r + permBase);   // attr[e0 + r + 8*hi, t]
        msg[r] += a * y[r];
      }
    }

    // scatter-add: out[dst[e]*16 + o] += msg (atomics hit L2-resident out)
#pragma unroll
    for (int r = 0; r < 8; ++r) {
      int drow = __shfl(dstn, r + permBase);     // dst[e0 + r + 8*hi]
      atomicAdd(&out[(long)drow * OUT_DIM + col], msg[r]);
    }
  }
}

extern "C" void kernel_launch(void* const* d_in, const int* in_sizes, int n_in,
                              void* d_out, int out_size, void* d_ws, size_t ws_size,
                              hipStream_t stream) {
  const float* x          = (const float*)d_in[0];
  const float* edge_attr  = (const float*)d_in[1];
  const float* root_w     = (const float*)d_in[2];
  const float* root_b     = (const float*)d_in[3];
  const float* edge_w     = (const float*)d_in[4];
  const float* edge_b     = (const float*)d_in[5];
  const long long* eidx   = (const long long*)d_in[6];
  float* out              = (float*)d_out;

  const int nodeTiles = N_NODES / 16;   // 6250 (exact)
  const int edgeTiles = N_EDGES / 16;   // 100000 (exact)

  // Kernel 1 initializes out; same-stream ordering guarantees it completes
  // before edge atomics begin.
  int rblocks = (nodeTiles + 7) / 8;    // 8 waves per 256-thread block
  ecconv_root_kernel<<<rblocks, 256, 0, stream>>>(x, root_w, root_b, out, nodeTiles);

  int eblocks = 1280;                   // 10240 waves, ~10 edge-tiles each
  ecconv_edge_kernel<<<eblocks, 256, 0, stream>>>(x, edge_attr, edge_w, edge_b,
                                                  eidx, out, edgeTiles);
}